// VariationalSAGEEncoder_11458972746376
// MI455X (gfx1250) — compile-verified
//
#include <hip/hip_runtime.h>
#include <hip/hip_bf16.h>

typedef __attribute__((ext_vector_type(2))) float v2f;
typedef __attribute__((ext_vector_type(8))) float v8f;

#define C_DIM 256   // K for every GEMM in this network

// ---------------------------------------------------------------------------
// Zero-fill workspace regions (ws is poisoned once, never re-poisoned).
// ---------------------------------------------------------------------------
__global__ void zero_f32(float* __restrict__ p, long long n) {
  long long i = (long long)blockIdx.x * blockDim.x + threadIdx.x;
  long long stride = (long long)gridDim.x * blockDim.x;
  for (; i < n; i += stride) p[i] = 0.0f;
}

// ---------------------------------------------------------------------------
// Weight transpose: Wt[n*K + k] = W[k*N + n]. Weights are tiny (<=256KB) and
// constant per launch; transposing once makes every GEMM B-operand load a
// contiguous b64 (two consecutive k per lane) instead of two strided b32s.
// ---------------------------------------------------------------------------
__global__ void transpose_w(const float* __restrict__ W, float* __restrict__ Wt,
                            int K, int N) {
  int t = blockIdx.x * blockDim.x + threadIdx.x;
  if (t >= K * N) return;
  int k = t / N;
  int n = t % N;
  Wt[n * K + k] = W[t];
}

// ---------------------------------------------------------------------------
// Edge scatter: one thread per (edge, 4-channel chunk). float4 gather from
// x[src], 4 global_atomic_add_f32 into agg[dst]; chunk 0 bumps the degree.
// agg fits in the 192MB L2 so atomic traffic is L2-resident.
// ---------------------------------------------------------------------------
__global__ void scatter_edges(const float* __restrict__ x,
                              const int* __restrict__ src,
                              const int* __restrict__ dst,
                              float* __restrict__ agg,
                              float* __restrict__ cnt,
                              int n_edges) {
  const int chunks = C_DIM / 4;  // 64
  long long t = (long long)blockIdx.x * blockDim.x + threadIdx.x;
  long long total = (long long)n_edges * chunks;
  if (t >= total) return;
  int e = (int)(t / chunks);
  int c = (int)(t % chunks) * 4;
  int s = src[e];
  int d = dst[e];
  const float4 v = *(const float4*)(x + (long long)s * C_DIM + c);
  float* a = agg + (long long)d * C_DIM + c;
  atomicAdd(a + 0, v.x);
  atomicAdd(a + 1, v.y);
  atomicAdd(a + 2, v.z);
  atomicAdd(a + 3, v.w);
  if (c == 0) atomicAdd(cnt + d, 1.0f);
}

// ---------------------------------------------------------------------------
// Fused dual-input GEMM with fp32 WMMA, fused scatter-mean scaling:
//   out[M,N] = (A1/deg)[M,K] @ W1 + A2[M,K] @ W2 + bias[N]   (opt. ReLU)
// W1/W2 passed TRANSPOSED as Wt[N][K]. Each wave computes a 16x64 strip:
// 4 accumulators, A fragment loaded once per k-step and reused 4x.
// Per-lane operand layout per CDNA5 ISA 7.12.2 (f32 16x16x4):
//   A: lane holds A[m0+(lane&15)][k + 2*(lane>>4) + {0,1}]   -> one b64 load
//   B: lane holds Wt[col][k + 2*(lane>>4) + {0,1}]           -> one b64 load
//   C/D: 8 VGPRs, row = r + 8*(lane>>4), col = n0+(lane&15)
// M multiple of 16, N multiple of 64, K=256 at every call -> EXEC all-1s and
// the tile guard is wave-uniform (no divergence around WMMA).
// ---------------------------------------------------------------------------
template <int RELU>
__global__ __launch_bounds__(256) void gemm_dual_wmma(
    const float* __restrict__ A1, const float* __restrict__ Wt1,
    const float* __restrict__ cnt1,  // degree counts for A1 (scatter-mean)
    const float* __restrict__ A2, const float* __restrict__ Wt2,
    const float* __restrict__ bias, float* __restrict__ out,
    int M, int N) {
  constexpr int K = C_DIM;
  const int lane = threadIdx.x & 31;
  const int wave = threadIdx.x >> 5;
  const int nstrips = N >> 6;                       // 16x64 strips per row band
  const int tile = blockIdx.x * 8 + wave;
  if (tile >= (M >> 4) * nstrips) return;           // wave-uniform guard

  const int m0 = (tile / nstrips) << 4;
  const int n0 = (tile % nstrips) << 6;

  const int half = lane >> 4;   // 0 or 1
  const int l15  = lane & 15;
  const int row  = m0 + l15;    // A row for this lane
  const int cb   = n0 + l15;    // base column for this lane
  const int koff = half << 1;   // per-lane K sub-offset: 0 or 2

  // scatter-mean scale for the aggregate path (cnt==0 rows -> mean 0 anyway)
  const float s1 = 1.0f / fmaxf(cnt1[row], 1.0f);

  v8f acc[4];
#pragma unroll
  for (int j = 0; j < 4; ++j) acc[j] = (v8f){0.f,0.f,0.f,0.f,0.f,0.f,0.f,0.f};

  const float* a1p = A1 + (long long)row * K + koff;
  const float* a2p = A2 + (long long)row * K + koff;
  const float* w1p = Wt1 + (long long)cb * K + koff;   // +16*K per strip col
  const float* w2p = Wt2 + (long long)cb * K + koff;

  // Path 1: mean-aggregate @ Wl
  for (int k = 0; k < K; k += 4) {
    v2f a = *(const v2f*)(a1p + k);
    a = a * s1;
#pragma unroll
    for (int j = 0; j < 4; ++j) {
      v2f b = *(const v2f*)(w1p + (long long)j * 16 * K + k);
      acc[j] = __builtin_amdgcn_wmma_f32_16x16x4_f32(false, a, false, b,
                                                     (short)0, acc[j],
                                                     false, false);
    }
  }
  // Path 2: x_dst @ Wr
  for (int k = 0; k < K; k += 4) {
    v2f a = *(const v2f*)(a2p + k);
#pragma unroll
    for (int j = 0; j < 4; ++j) {
      v2f b = *(const v2f*)(w2p + (long long)j * 16 * K + k);
      acc[j] = __builtin_amdgcn_wmma_f32_16x16x4_f32(false, a, false, b,
                                                     (short)0, acc[j],
                                                     false, false);
    }
  }

#pragma unroll
  for (int j = 0; j < 4; ++j) {
    const int col = cb + j * 16;
    const float bv = bias[col];
#pragma unroll
    for (int r = 0; r < 8; ++r) {
      float v = acc[j][r] + bv;
      if (RELU) v = fmaxf(v, 0.f);
      out[(long long)(m0 + r + half * 8) * N + col] = v;
    }
  }
}

// ---------------------------------------------------------------------------
// Host-side orchestration
// ---------------------------------------------------------------------------
static inline int cdiv(long long a, long long b) { return (int)((a + b - 1) / b); }

extern "C" void kernel_launch(void* const* d_in, const int* in_sizes, int n_in,
                              void* d_out, int out_size, void* d_ws, size_t ws_size,
                              hipStream_t stream) {
  (void)n_in; (void)out_size; (void)ws_size;

  const int N1 = 40000, N2 = 20000, N3 = 10000;
  const int OUT_C = 64;

  const float* x   = (const float*)d_in[0];
  const int*   ei1 = (const int*)d_in[1];
  const int*   ei2 = (const int*)d_in[2];
  const int*   ei3 = (const int*)d_in[3];
  const float* Wl1 = (const float*)d_in[4];
  const float* bl1 = (const float*)d_in[5];
  const float* Wr1 = (const float*)d_in[6];
  const float* Wl2 = (const float*)d_in[7];
  const float* bl2 = (const float*)d_in[8];
  const float* Wr2 = (const float*)d_in[9];
  const float* Wl3 = (const float*)d_in[10];
  const float* bl3 = (const float*)d_in[11];
  const float* Wr3 = (const float*)d_in[12];
  const float* Wls = (const float*)d_in[13];
  const float* bls = (const float*)d_in[14];
  const float* Wrs = (const float*)d_in[15];

  const int E1 = in_sizes[1] / 2;
  const int E2 = in_sizes[2] / 2;
  const int E3 = in_sizes[3] / 2;

  // Workspace layout (floats)
  float* h1  = (float*)d_ws;                        // 40000*256
  float* h2  = h1 + (long long)N1 * C_DIM;          // 20000*256
  float* agg = h2 + (long long)N2 * C_DIM;          // 40000*256 (reused)
  float* cnt = agg + (long long)N1 * C_DIM;         // 40000
  float* wt  = cnt + N1;                            // transposed weights
  float* Wl1t = wt;                                 // 256*256
  float* Wr1t = Wl1t + C_DIM * C_DIM;
  float* Wl2t = Wr1t + C_DIM * C_DIM;
  float* Wr2t = Wl2t + C_DIM * C_DIM;
  float* Wl3t = Wr2t + C_DIM * C_DIM;               // 256*64 each from here
  float* Wr3t = Wl3t + C_DIM * OUT_C;
  float* Wlst = Wr3t + C_DIM * OUT_C;
  float* Wrst = Wlst + C_DIM * OUT_C;

  float* mu     = (float*)d_out;                    // 10000*64
  float* logstd = mu + (long long)N3 * OUT_C;       // 10000*64

  const int ZB = 256, SB = 256, GB = 256, TB = 256;

  // ---- Transpose all weights up front (constant per launch, ~1.3MB) -------
  {
    int nbig = C_DIM * C_DIM, nsml = C_DIM * OUT_C;
    transpose_w<<<cdiv(nbig, TB), TB, 0, stream>>>(Wl1, Wl1t, C_DIM, C_DIM);
    transpose_w<<<cdiv(nbig, TB), TB, 0, stream>>>(Wr1, Wr1t, C_DIM, C_DIM);
    transpose_w<<<cdiv(nbig, TB), TB, 0, stream>>>(Wl2, Wl2t, C_DIM, C_DIM);
    transpose_w<<<cdiv(nbig, TB), TB, 0, stream>>>(Wr2, Wr2t, C_DIM, C_DIM);
    transpose_w<<<cdiv(nsml, TB), TB, 0, stream>>>(Wl3, Wl3t, C_DIM, OUT_C);
    transpose_w<<<cdiv(nsml, TB), TB, 0, stream>>>(Wr3, Wr3t, C_DIM, OUT_C);
    transpose_w<<<cdiv(nsml, TB), TB, 0, stream>>>(Wls, Wlst, C_DIM, OUT_C);
    transpose_w<<<cdiv(nsml, TB), TB, 0, stream>>>(Wrs, Wrst, C_DIM, OUT_C);
  }

  // ---------------- Layer 1: mean-aggregate x over ei1 -> h1 ----------------
  {
    long long na = (long long)N1 * C_DIM;
    zero_f32<<<cdiv(na, ZB), ZB, 0, stream>>>(agg, na);
    zero_f32<<<cdiv(N1, ZB), ZB, 0, stream>>>(cnt, N1);
    long long work = (long long)E1 * (C_DIM / 4);
    scatter_edges<<<cdiv(work, SB), SB, 0, stream>>>(x, ei1, ei1 + E1, agg, cnt, E1);
    int tiles = (N1 / 16) * (C_DIM / 64);
    gemm_dual_wmma<1><<<cdiv(tiles, 8), GB, 0, stream>>>(
        agg, Wl1t, cnt, x, Wr1t, bl1, h1, N1, C_DIM);
  }

  // ---------------- Layer 2: mean-aggregate h1 over ei2 -> h2 ---------------
  {
    long long na = (long long)N2 * C_DIM;
    zero_f32<<<cdiv(na, ZB), ZB, 0, stream>>>(agg, na);
    zero_f32<<<cdiv(N2, ZB), ZB, 0, stream>>>(cnt, N2);
    long long work = (long long)E2 * (C_DIM / 4);
    scatter_edges<<<cdiv(work, SB), SB, 0, stream>>>(h1, ei2, ei2 + E2, agg, cnt, E2);
    int tiles = (N2 / 16) * (C_DIM / 64);
    gemm_dual_wmma<1><<<cdiv(tiles, 8), GB, 0, stream>>>(
        agg, Wl2t, cnt, h1, Wr2t, bl2, h2, N2, C_DIM);
  }

  // ------- Layer 3: aggregate ONCE over ei3, shared by mu & logstd ----------
  {
    long long na = (long long)N3 * C_DIM;
    zero_f32<<<cdiv(na, ZB), ZB, 0, stream>>>(agg, na);
    zero_f32<<<cdiv(N3, ZB), ZB, 0, stream>>>(cnt, N3);
    long long work = (long long)E3 * (C_DIM / 4);
    scatter_edges<<<cdiv(work, SB), SB, 0, stream>>>(h2, ei3, ei3 + E3, agg, cnt, E3);

    int tiles = (N3 / 16) * (OUT_C / 64);
    gemm_dual_wmma<0><<<cdiv(tiles, 8), GB, 0, stream>>>(
        agg, Wl3t, cnt, h2, Wr3t, bl3, mu, N3, OUT_C);
    gemm_dual_wmma<0><<<cdiv(tiles, 8), GB, 0, stream>>>(
        agg, Wlst, cnt, h2, Wrst, bls, logstd, N3, OUT_C);
  }
}